// GraphSAGELayer_51299089384083
// MI455X (gfx1250) — compile-verified
//
#include <hip/hip_runtime.h>
#include <math.h>

#define N_NODES 100000
#define FEATS   256
#define LDSTR   260   // padded LDS row stride (floats): breaks 16-way bank conflicts

typedef float v2f __attribute__((ext_vector_type(2)));
typedef float v8f __attribute__((ext_vector_type(8)));

// ---------------- zero workspace ----------------
__global__ void zero_f4_kernel(float4* __restrict__ p, int n4) {
    float4 z = {0.f, 0.f, 0.f, 0.f};
    for (int i = blockIdx.x * blockDim.x + threadIdx.x; i < n4;
         i += gridDim.x * blockDim.x)
        p[i] = z;
}

// ---------------- edge aggregation: wave-per-edge scatter-add ----------------
__global__ void edge_agg_kernel(const float* __restrict__ x,
                                const long long* __restrict__ ei,   // [2,E] int64
                                const float* __restrict__ ew,
                                float* __restrict__ num,            // [N,256] accum
                                float* __restrict__ wsum,           // [N] accum
                                int E) {
    int lane = threadIdx.x & 31;
    int wid  = (blockIdx.x * blockDim.x + threadIdx.x) >> 5;
    int nw   = (gridDim.x * blockDim.x) >> 5;
    for (int e = wid; e < E; e += nw) {
        long long row = ei[e];
        long long col = ei[(long long)E + e];
        float w = ew[e];
        const float4* xp = (const float4*)(x + row * (long long)FEATS) + lane * 2;
        float4 a = xp[0];
        float4 b = xp[1];
        float* np = num + col * (long long)FEATS + lane * 8;
        unsafeAtomicAdd(np + 0, a.x * w);
        unsafeAtomicAdd(np + 1, a.y * w);
        unsafeAtomicAdd(np + 2, a.z * w);
        unsafeAtomicAdd(np + 3, a.w * w);
        unsafeAtomicAdd(np + 4, b.x * w);
        unsafeAtomicAdd(np + 5, b.y * w);
        unsafeAtomicAdd(np + 6, b.z * w);
        unsafeAtomicAdd(np + 7, b.w * w);
        if (lane == 0) unsafeAtomicAdd(wsum + col, w);
    }
}

// ---------------- fused GEMM (fp32 WMMA) + bias + L2 normalize ----------------
// Block: 512 threads = 16 waves. Block handles 16 node-rows; wave w handles
// output columns [16w, 16w+16). K loop = 256 in steps of 4 via
// v_wmma_f32_16x16x4_f32, accumulating x@Ws^T + neigh@Wn^T in one f32 acc.
__global__ __launch_bounds__(512)
void sage_gemm_kernel(const float* __restrict__ x,
                      const float* __restrict__ num,
                      const float* __restrict__ wsum,
                      const float* __restrict__ Ws,    // [256,256] row-major [out][in]
                      const float* __restrict__ Wn,
                      const float* __restrict__ bias,  // [256]
                      float* __restrict__ out) {
    __shared__ float shx[16 * LDSTR];
    __shared__ float shn[16 * LDSTR];
    __shared__ float shnrm[16];

    const int t    = threadIdx.x;
    const int row0 = blockIdx.x * 16;

    // ---- stage x tile and neigh = num/max(wsum,eps) tile into LDS (float4) ----
    for (int v = t; v < 16 * 64; v += 512) {
        int r = v >> 6, c = v & 63;
        size_t g = (size_t)(row0 + r) * FEATS + c * 4;
        float4 xv = *(const float4*)(x + g);
        *(float4*)(shx + r * LDSTR + c * 4) = xv;
        float inv = 1.0f / fmaxf(wsum[row0 + r], 1e-8f);
        float4 nv = *(const float4*)(num + g);
        nv.x *= inv; nv.y *= inv; nv.z *= inv; nv.w *= inv;
        *(float4*)(shn + r * LDSTR + c * 4) = nv;
    }
    __syncthreads();

    const int wv   = t >> 5;          // wave id 0..15 -> column tile
    const int lane = t & 31;
    const int j0   = wv * 16;
    const int m    = lane & 15;       // row-in-tile (A) / col-in-tile (B,D)
    const int koff = (lane >> 4) * 2; // lanes 16-31 hold K+2,K+3

    const float* pax = shx + m * LDSTR + koff;
    const float* pan = shn + m * LDSTR + koff;
    const float* pbs = Ws + (size_t)(j0 + m) * FEATS + koff;  // B[k][j] = W[j][k]
    const float* pbn = Wn + (size_t)(j0 + m) * FEATS + koff;

    v8f acc = {0.f, 0.f, 0.f, 0.f, 0.f, 0.f, 0.f, 0.f};
    #pragma unroll 8
    for (int k0 = 0; k0 < FEATS; k0 += 4) {
        v2f ax = *(const v2f*)(pax + k0);
        v2f bs = *(const v2f*)(pbs + k0);
        acc = __builtin_amdgcn_wmma_f32_16x16x4_f32(false, ax, false, bs,
                                                    (short)0, acc, false, false);
        v2f an = *(const v2f*)(pan + k0);
        v2f bn = *(const v2f*)(pbn + k0);
        acc = __builtin_amdgcn_wmma_f32_16x16x4_f32(false, an, false, bn,
                                                    (short)0, acc, false, false);
    }

    float bv = bias[j0 + m];
    __syncthreads();   // everyone done reading shx/shn; reuse shx as out staging

    // D layout: VGPR r -> row r (lanes 0-15) / row r+8 (lanes 16-31), col j0+m
    const int rbase = (lane >> 4) * 8;
    #pragma unroll
    for (int r = 0; r < 8; ++r)
        shx[(rbase + r) * LDSTR + j0 + m] = acc[r] + bv;
    __syncthreads();

    // ---- per-row L2 norm: wave wv reduces row wv ----
    {
        float s = 0.f;
        #pragma unroll
        for (int c = lane; c < FEATS; c += 32) {
            float v = shx[wv * LDSTR + c];
            s += v * v;
        }
        #pragma unroll
        for (int off = 16; off > 0; off >>= 1)
            s += __shfl_xor(s, off, 32);
        if (lane == 0)
            shnrm[wv] = 1.0f / fmaxf(sqrtf(s), 1e-12f);
    }
    __syncthreads();

    // ---- normalized float4 stores ----
    for (int v = t; v < 16 * 64; v += 512) {
        int r = v >> 6, c = v & 63;
        float inv = shnrm[r];
        float4 o = *(const float4*)(shx + r * LDSTR + c * 4);
        o.x *= inv; o.y *= inv; o.z *= inv; o.w *= inv;
        *(float4*)(out + (size_t)(row0 + r) * FEATS + c * 4) = o;
    }
}

extern "C" void kernel_launch(void* const* d_in, const int* in_sizes, int n_in,
                              void* d_out, int out_size, void* d_ws, size_t ws_size,
                              hipStream_t stream) {
    (void)n_in; (void)out_size; (void)ws_size;
    const float*     x    = (const float*)d_in[0];
    const long long* ei   = (const long long*)d_in[1];   // int64 [2,E]
    const float*     ew   = (const float*)d_in[2];
    const float*     Ws   = (const float*)d_in[3];
    const float*     Wn   = (const float*)d_in[4];
    const float*     bias = (const float*)d_in[5];
    float*           out  = (float*)d_out;
    const int E = in_sizes[2];

    float* num  = (float*)d_ws;                       // [N,256]
    float* wsum = num + (size_t)N_NODES * FEATS;      // [N]

    zero_f4_kernel<<<2048, 256, 0, stream>>>((float4*)num, (N_NODES * FEATS) / 4);
    zero_f4_kernel<<<64,   256, 0, stream>>>((float4*)wsum, N_NODES / 4);
    edge_agg_kernel<<<2048, 256, 0, stream>>>(x, ei, ew, num, wsum, E);
    sage_gemm_kernel<<<N_NODES / 16, 512, 0, stream>>>(x, num, wsum, Ws, Wn, bias, out);
}